// CannyAppLayer_22728966930729
// MI455X (gfx1250) — compile-verified
//
#include <hip/hip_runtime.h>
#include <stdint.h>

// ---------------------------------------------------------------------------
// Problem constants: x is (B=32, H=512, W=512, C=3) float32, NHWC.
// ---------------------------------------------------------------------------
constexpr int Bn  = 32;
constexpr int Hn  = 512;
constexpr int Wn  = 512;
constexpr int Cn  = 3;
constexpr int HWn = Hn * Wn;            // 262144
constexpr int PL  = Bn * Cn;            // 96 planes
constexpr int Ntot = Bn * HWn * Cn;     // 25165824

// ---------------------------------------------------------------------------
// Workspace layout (≈50.7 MB)
// ---------------------------------------------------------------------------
constexpr size_t OFF_MAX  = 0;                                   // float
constexpr size_t OFF_HIST = 256;                                 // u32 [96][256]
constexpr size_t HISTSZ   = (size_t)PL * 256 * 4;
constexpr size_t OFF_W0   = OFF_HIST + HISTSZ;                   // float [96][256]
constexpr size_t OFF_S0   = OFF_W0 + (size_t)PL * 256 * 4;       // float [96][256]
constexpr size_t OFF_HIGH = OFF_S0 + (size_t)PL * 256 * 4;       // float [96]
constexpr size_t OFF_G    = (OFF_HIGH + PL * 4 + 255) & ~(size_t)255;  // u8 planes
constexpr size_t OFF_MASK = OFF_G + (size_t)PL * HWn;            // u8 planes
constexpr int INIT_WORDS  = (int)((OFF_HIST + HISTSZ) / 4);

// ---------------------------------------------------------------------------
// CDNA5 feature probes (compile-only loop: fall back gracefully, histogram
// tells us whether the gfx1250 paths were taken).
// ---------------------------------------------------------------------------
typedef __attribute__((ext_vector_type(2))) float v2f;
typedef __attribute__((ext_vector_type(8))) float v8f;

#if defined(__has_builtin)
#  if __has_builtin(__builtin_amdgcn_wmma_f32_16x16x4_f32)
#    define HAVE_WMMA4 1
#  endif
#  if __has_builtin(__builtin_amdgcn_global_load_async_to_lds_b32)
#    define HAVE_ASYNC 1
#  endif
#  if __has_builtin(__builtin_amdgcn_s_wait_asynccnt)
#    define HAVE_WAIT_ASYNC 1
#  endif
#endif

__device__ __forceinline__ void async_g2l_b32(const uint8_t* gsrc, void* ldst) {
#ifdef HAVE_ASYNC
  __builtin_amdgcn_global_load_async_to_lds_b32(
      (__attribute__((address_space(1))) int*)(gsrc),
      (__attribute__((address_space(3))) int*)(ldst),
      /*offset=*/0, /*cpol=*/0);
#else
  *(uint32_t*)ldst = *(const uint32_t*)gsrc;   // synchronous fallback
#endif
}

__device__ __forceinline__ void wait_async_all() {
#ifdef HAVE_ASYNC
#  ifdef HAVE_WAIT_ASYNC
  __builtin_amdgcn_s_wait_asynccnt(0);
#  else
  asm volatile("s_wait_asynccnt 0" ::: "memory");
#  endif
#endif
}

__device__ __forceinline__ int clampi(int v, int lo, int hi) {
  return v < lo ? lo : (v > hi ? hi : v);
}

// ---------------------------------------------------------------------------
// 0) zero the max slot + histograms (harness poisons ws; we must be idempotent)
// ---------------------------------------------------------------------------
__global__ void k_init(unsigned* ws) {
  int i = blockIdx.x * 256 + threadIdx.x;
  if (i < INIT_WORDS) ws[i] = 0u;
}

// ---------------------------------------------------------------------------
// 1) global max of x  (decides the *255 scaling, as in the reference)
// ---------------------------------------------------------------------------
__global__ void k_max(const float4* __restrict__ x4, float* __restrict__ maxout, int n4) {
  __shared__ float red[256];
  float m = 0.f;
  for (int i = blockIdx.x * blockDim.x + threadIdx.x; i < n4; i += gridDim.x * blockDim.x) {
    float4 v = x4[i];
    m = fmaxf(m, fmaxf(fmaxf(v.x, v.y), fmaxf(v.z, v.w)));
  }
  red[threadIdx.x] = m;
  __syncthreads();
  for (int s = 128; s > 0; s >>= 1) {
    if (threadIdx.x < s) red[threadIdx.x] = fmaxf(red[threadIdx.x], red[threadIdx.x + s]);
    __syncthreads();
  }
  if (threadIdx.x == 0) atomicMax((unsigned*)maxout, __float_as_uint(red[0]));  // x >= 0
}

// ---------------------------------------------------------------------------
// 2) quantize -> u8 planar g, per-plane 256-bin histogram via LDS atomics
// ---------------------------------------------------------------------------
__global__ void k_hist(const float* __restrict__ x, const float* __restrict__ maxp,
                       unsigned* __restrict__ hist, uint8_t* __restrict__ g) {
  __shared__ unsigned lh[3 * 256];
  int t = threadIdx.x;
  lh[t] = 0u; lh[t + 256] = 0u; lh[t + 512] = 0u;
  __syncthreads();

  float scale = (*maxp < 1.1f) ? 255.f : 1.f;
  int hw = blockIdx.x * 256 + t;
  int b  = blockIdx.y;
  const float* px = x + ((size_t)b * HWn + hw) * 3;
#pragma unroll
  for (int c = 0; c < 3; ++c) {
    float v = floorf(fminf(fmaxf(px[c] * scale, 0.f), 255.f));
    int gi = (int)v;
    atomicAdd(&lh[c * 256 + gi], 1u);
    g[((size_t)(b * 3 + c)) * HWn + hw] = (uint8_t)gi;
  }
  __syncthreads();
#pragma unroll
  for (int c = 0; c < 3; ++c)
    atomicAdd(&hist[(b * 3 + c) * 256 + t], lh[c * 256 + t]);
}

// ---------------------------------------------------------------------------
// 3) Otsu cumulative sums as a real GEMM:
//    W0 = Hist(96x256) x U,  S0 = (Hist .* bins) x U,  U[i][j] = (i<=j)
//    One wave per 16x16 output tile, K-loop of 64 x V_WMMA_F32_16X16X4_F32.
//    Layouts per ISA 7.12.2: A 16x4 (VGPR0:K0/K2, VGPR1:K1/K3 per lane-half),
//    C/D 16x16 (VGPRv: M=v lanes0-15, M=v+8 lanes16-31).
// ---------------------------------------------------------------------------
#ifdef HAVE_WMMA4
__global__ void k_cumsum(const unsigned* __restrict__ hist,
                         float* __restrict__ w0, float* __restrict__ s0) {
  int lane  = threadIdx.x;          // 0..31, wave32
  int row   = lane & 15;            // A row (M) / B col (N)
  int half  = lane >> 4;            // K sub-pair selector
  int mBase = blockIdx.y * 16;      // plane tile
  int jBase = blockIdx.x * 16;      // bin tile
  int j     = jBase + row;

  v8f cw = {0.f, 0.f, 0.f, 0.f, 0.f, 0.f, 0.f, 0.f};
  v8f cs = {0.f, 0.f, 0.f, 0.f, 0.f, 0.f, 0.f, 0.f};

  for (int k4 = 0; k4 < 64; ++k4) {
    int k0 = k4 * 4 + half * 2;
    float h0 = (float)hist[(mBase + row) * 256 + k0];
    float h1 = (float)hist[(mBase + row) * 256 + k0 + 1];
    v2f a  = {h0, h1};
    v2f as = {h0 * (float)k0, h1 * (float)(k0 + 1)};
    v2f bb = {(k0 <= j) ? 1.f : 0.f, (k0 + 1 <= j) ? 1.f : 0.f};
    cw = __builtin_amdgcn_wmma_f32_16x16x4_f32(false, a,  false, bb, (short)0, cw, false, false);
    cs = __builtin_amdgcn_wmma_f32_16x16x4_f32(false, as, false, bb, (short)0, cs, false, false);
  }
  int mAdd = half * 8;
#pragma unroll
  for (int v = 0; v < 8; ++v) {
    w0[(size_t)(mBase + v + mAdd) * 256 + jBase + row] = cw[v];
    s0[(size_t)(mBase + v + mAdd) * 256 + jBase + row] = cs[v];
  }
}
#endif

// ---------------------------------------------------------------------------
// 4) Otsu threshold per plane: between-class variance + first-index argmax
// ---------------------------------------------------------------------------
__global__ void k_otsu(const unsigned* __restrict__ hist,
                       const float* __restrict__ w0g, const float* __restrict__ s0g,
                       float* __restrict__ high) {
  __shared__ float sw[256], ss[256];
  __shared__ float bv[256];
  __shared__ int   bi[256];
  int p = blockIdx.x, t = threadIdx.x;

#ifdef HAVE_WMMA4
  (void)hist;
  sw[t] = w0g[p * 256 + t];
  ss[t] = s0g[p * 256 + t];
  __syncthreads();
#else
  (void)w0g; (void)s0g;
  float h = (float)hist[p * 256 + t];
  sw[t] = h;
  ss[t] = h * (float)t;
  __syncthreads();
  for (int off = 1; off < 256; off <<= 1) {          // Hillis-Steele inclusive scan
    float aw = sw[t], as_ = ss[t];
    float bw = (t >= off) ? sw[t - off] : 0.f;
    float bs = (t >= off) ? ss[t - off] : 0.f;
    __syncthreads();
    sw[t] = aw + bw;
    ss[t] = as_ + bs;
    __syncthreads();
  }
#endif

  float total = sw[255], sT = ss[255];
  float w0v = sw[t], s0v = ss[t];
  float w1  = total - w0v;
  float mu0 = s0v / fmaxf(w0v, 1.f);
  float mu1 = (sT - s0v) / fmaxf(w1, 1.f);
  float d   = mu0 - mu1;
  float sb  = (w0v > 0.f && w1 > 0.f) ? w0v * w1 * d * d : -1.f;
  bv[t] = sb; bi[t] = t;
  __syncthreads();
  for (int s = 128; s > 0; s >>= 1) {
    if (t < s) {
      float ob = bv[t + s]; int oi = bi[t + s];
      if (ob > bv[t] || (ob == bv[t] && oi < bi[t])) { bv[t] = ob; bi[t] = oi; }
    }
    __syncthreads();
  }
  if (t == 0) high[p] = (float)bi[0];
}

// ---------------------------------------------------------------------------
// 5) Sobel + NMS + double-threshold for one 32x32 tile per block.
//    g tile (36 rows x 40 cols, halo 2, x padded to dword-aligned window)
//    staged via ASYNC global->LDS b32 DMA, then s_wait_asynccnt.
// ---------------------------------------------------------------------------
__global__ void k_sobel(const uint8_t* __restrict__ g, const float* __restrict__ high,
                        uint8_t* __restrict__ mask) {
  __shared__ uint8_t gt[36][40];
  __shared__ float   mg[34][36];
  int t   = threadIdx.x;
  int TX0 = blockIdx.x * 32, TY0 = blockIdx.y * 32, p = blockIdx.z;
  const uint8_t* gp = g + (size_t)p * HWn;

  // async DMA: 36 rows x 10 dwords, y clamped per-row (edge padding),
  // x window [TX0-4, TX0+36) clamped at dword granularity
  for (int d = t; d < 360; d += 256) {
    int r  = d / 10, dw = d % 10;
    int y  = clampi(TY0 - 2 + r, 0, Hn - 1);
    int gx = clampi(TX0 - 4 + dw * 4, 0, Wn - 4);
    async_g2l_b32(gp + (size_t)y * Wn + gx, &gt[r][dw * 4]);
  }
  wait_async_all();
  __syncthreads();

  // x edge-replication fixups (only border tiles)
  if (TX0 == 0) {
    for (int i = t; i < 36 * 4; i += 256) { int r = i >> 2, c = i & 3; gt[r][c] = gt[r][4]; }
  }
  if (TX0 == Wn - 32) {
    for (int i = t; i < 36 * 4; i += 256) { int r = i >> 2, c = 36 + (i & 3); gt[r][c] = gt[r][35]; }
  }
  __syncthreads();

  // |gx|+|gy| magnitude on (34x34) region (zero outside image, per jnp.pad)
  for (int i = t; i < 34 * 34; i += 256) {
    int r = i / 34, c = i % 34;
    int y = TY0 - 1 + r, x = TX0 - 1 + c;
    float m = 0.f;
    if (y >= 0 && y < Hn && x >= 0 && x < Wn) {
      int rr = r + 1, cc = c + 3;
      float p00 = gt[rr-1][cc-1], p01 = gt[rr-1][cc], p02 = gt[rr-1][cc+1];
      float p10 = gt[rr  ][cc-1],                     p12 = gt[rr  ][cc+1];
      float p20 = gt[rr+1][cc-1], p21 = gt[rr+1][cc], p22 = gt[rr+1][cc+1];
      float gxv = (p02 + 2.f*p12 + p22) - (p00 + 2.f*p10 + p20);
      float gyv = (p20 + 2.f*p21 + p22) - (p00 + 2.f*p01 + p02);
      m = fabsf(gxv) + fabsf(gyv);
    }
    mg[r][c] = m;
  }
  __syncthreads();

  float hi = high[p], lo = 0.33f * hi;
  for (int i = t; i < 1024; i += 256) {
    int r = i >> 5, c = i & 31;
    int rr = r + 2, cc = c + 4;
    float p00 = gt[rr-1][cc-1], p01 = gt[rr-1][cc], p02 = gt[rr-1][cc+1];
    float p10 = gt[rr  ][cc-1],                     p12 = gt[rr  ][cc+1];
    float p20 = gt[rr+1][cc-1], p21 = gt[rr+1][cc], p22 = gt[rr+1][cc+1];
    float gxv = (p02 + 2.f*p12 + p22) - (p00 + 2.f*p10 + p20);
    float gyv = (p20 + 2.f*p21 + p22) - (p00 + 2.f*p01 + p02);
    float m   = mg[r+1][c+1];
    float ax  = fabsf(gxv), ay = fabsf(gyv);
    float n1, n2;
    if (ay < 0.41421356f * ax)      { n1 = mg[r+1][c+2]; n2 = mg[r+1][c  ]; }  // 0°  E/W
    else if (ay > 2.41421356f * ax) { n1 = mg[r  ][c+1]; n2 = mg[r+2][c+1]; }  // 90° N/S
    else if (gxv * gyv > 0.f)       { n1 = mg[r  ][c+2]; n2 = mg[r+2][c  ]; }  // 45° NE/SW
    else                            { n1 = mg[r  ][c  ]; n2 = mg[r+2][c+2]; }  // 135° NW/SE
    float nms = (m >= n1 && m >= n2) ? m : 0.f;
    uint8_t mk = (nms > hi) ? (uint8_t)2 : ((nms > lo) ? (uint8_t)1 : (uint8_t)0);
    mask[(size_t)p * HWn + (size_t)(TY0 + r) * Wn + (TX0 + c)] = mk;
  }
}

// ---------------------------------------------------------------------------
// 6) hysteresis pass: local fixed point within a 32x32 tile (+1 halo) in LDS.
//    Promotion 1->2 is monotone, so cross-block races are benign; fixed 12
//    global passes (graph capture forbids data-dependent launch counts).
// ---------------------------------------------------------------------------
__global__ void k_hyst(uint8_t* __restrict__ mask) {
  __shared__ uint8_t mt[34][36];
  __shared__ int changed;
  int t   = threadIdx.x;
  int TX0 = blockIdx.x * 32, TY0 = blockIdx.y * 32, p = blockIdx.z;
  uint8_t* mp = mask + (size_t)p * HWn;

  for (int i = t; i < 34 * 34; i += 256) {
    int r = i / 34, c = i % 34;
    int y = TY0 - 1 + r, x = TX0 - 1 + c;
    mt[r][c] = (y >= 0 && y < Hn && x >= 0 && x < Wn) ? mp[(size_t)y * Wn + x] : (uint8_t)0;
  }
  __syncthreads();

  for (int it = 0; it < 16; ++it) {
    if (t == 0) changed = 0;
    __syncthreads();
    for (int i = t; i < 1024; i += 256) {
      int r = i / 32 + 1, c = i % 32 + 1;
      if (mt[r][c] == 1) {
        int s = (mt[r-1][c-1] == 2) | (mt[r-1][c] == 2) | (mt[r-1][c+1] == 2) |
                (mt[r  ][c-1] == 2) |                     (mt[r  ][c+1] == 2) |
                (mt[r+1][c-1] == 2) | (mt[r+1][c] == 2) | (mt[r+1][c+1] == 2);
        if (s) { mt[r][c] = 2; changed = 1; }
      }
    }
    __syncthreads();
    int ch = changed;
    __syncthreads();
    if (!ch) break;
  }

  for (int i = t; i < 1024; i += 256) {
    int r = i / 32 + 1, c = i % 32 + 1;
    mp[(size_t)(TY0 + r - 1) * Wn + (TX0 + c - 1)] = mt[r][c];
  }
}

// ---------------------------------------------------------------------------
// 7) finalize: planar mask -> NHWC float edges, nontemporal stores
// ---------------------------------------------------------------------------
__global__ void k_final(const uint8_t* __restrict__ mask, float* __restrict__ out) {
  int i  = blockIdx.x * 256 + threadIdx.x;     // grid sized exactly to Ntot
  int c  = i % 3;
  int hw = (i / 3) % HWn;
  int b  = i / (3 * HWn);
  float v = (mask[((size_t)(b * 3 + c)) * HWn + hw] == 2) ? 1.f : 0.f;
  __builtin_nontemporal_store(v, &out[i]);
}

// ---------------------------------------------------------------------------
extern "C" void kernel_launch(void* const* d_in, const int* in_sizes, int n_in,
                              void* d_out, int out_size, void* d_ws, size_t ws_size,
                              hipStream_t stream) {
  (void)in_sizes; (void)n_in; (void)out_size; (void)ws_size;
  const float* x = (const float*)d_in[0];
  uint8_t* ws = (uint8_t*)d_ws;

  float*    maxp = (float*)(ws + OFF_MAX);
  unsigned* hist = (unsigned*)(ws + OFF_HIST);
  float*    w0   = (float*)(ws + OFF_W0);
  float*    s0   = (float*)(ws + OFF_S0);
  float*    high = (float*)(ws + OFF_HIGH);
  uint8_t*  g    = ws + OFF_G;
  uint8_t*  mask = ws + OFF_MASK;
  float*    out  = (float*)d_out;

  k_init<<<(INIT_WORDS + 255) / 256, 256, 0, stream>>>((unsigned*)ws);
  k_max<<<1024, 256, 0, stream>>>((const float4*)x, maxp, Ntot / 4);
  k_hist<<<dim3(HWn / 256, Bn), 256, 0, stream>>>(x, maxp, hist, g);
#ifdef HAVE_WMMA4
  k_cumsum<<<dim3(16, 6), 32, 0, stream>>>(hist, w0, s0);
#endif
  k_otsu<<<PL, 256, 0, stream>>>(hist, w0, s0, high);
  k_sobel<<<dim3(Wn / 32, Hn / 32, PL), 256, 0, stream>>>(g, high, mask);
  for (int pass = 0; pass < 12; ++pass)
    k_hyst<<<dim3(Wn / 32, Hn / 32, PL), 256, 0, stream>>>(mask);
  k_final<<<Ntot / 256, 256, 0, stream>>>(mask, out);
}